// FirstOrderWaveletTransform_8761733284634
// MI455X (gfx1250) — compile-verified
//
#include <hip/hip_runtime.h>

// Haar DWT2 level-1, x:(16,32,512,512) f32 -> (LL, cH+cV+cD), each (16,32,256,256).
// Pure bandwidth-bound streaming kernel: 512 MiB read + 256 MiB write, ~33 us floor
// at 23.3 TB/s. Uses 128-bit non-temporal loads/stores (data touched once; input
// exceeds the 192 MB L2, so NT avoids cache thrash).

#define NIMG 512   // 16 * 32
#define HH   512
#define WW   512
#define OH   256
#define OW   256

// Native clang vector (HIP's float4 is a struct and is rejected by the
// nontemporal builtins).
typedef __attribute__((ext_vector_type(4))) float v4f;

__global__ __launch_bounds__(256)
void haar_dwt2_kernel(const float* __restrict__ x,
                      float* __restrict__ ll,
                      float* __restrict__ hs) {
    // One thread per 4 consecutive output columns of one output row.
    // tid layout: [img(9b) | oh(8b) | owg(6b)]  -> 512*256*64 = 8,388,608 threads.
    const int tid = blockIdx.x * blockDim.x + threadIdx.x;
    const int owg = tid & 63;          // which group of 4 output columns
    const int oh  = (tid >> 6) & 255;  // output row
    const int img = tid >> 14;         // (b,c) image index

    const size_t in_base = ((size_t)img * HH + 2u * (size_t)oh) * WW + (size_t)owg * 8u;
    const v4f* r0 = reinterpret_cast<const v4f*>(x + in_base);       // row 2h
    const v4f* r1 = reinterpret_cast<const v4f*>(x + in_base + WW);  // row 2h+1

    // 64 bytes in per thread, contiguous 1 KiB per wave per row: global_load_b128 NT.
    v4f a0 = __builtin_nontemporal_load(r0);
    v4f a1 = __builtin_nontemporal_load(r0 + 1);
    v4f b0 = __builtin_nontemporal_load(r1);
    v4f b1 = __builtin_nontemporal_load(r1 + 1);

    // LL = (x00 + x01 + x10 + x11) * 0.5  (same left-to-right association as ref)
    // Hsum = cH + cV + cD == 2*x00 - LL   (exact in real arithmetic)
    v4f vll, vhs;
    vll.x = (a0.x + a0.y + b0.x + b0.y) * 0.5f;
    vll.y = (a0.z + a0.w + b0.z + b0.w) * 0.5f;
    vll.z = (a1.x + a1.y + b1.x + b1.y) * 0.5f;
    vll.w = (a1.z + a1.w + b1.z + b1.w) * 0.5f;

    vhs.x = 2.0f * a0.x - vll.x;
    vhs.y = 2.0f * a0.z - vll.y;
    vhs.z = 2.0f * a1.x - vll.z;
    vhs.w = 2.0f * a1.z - vll.w;

    const size_t obase = ((size_t)img * OH + (size_t)oh) * OW + (size_t)owg * 4u;
    // 32 bytes out per thread, contiguous 512 B per wave per array: global_store_b128 NT.
    __builtin_nontemporal_store(vll, reinterpret_cast<v4f*>(ll + obase));
    __builtin_nontemporal_store(vhs, reinterpret_cast<v4f*>(hs + obase));
}

extern "C" void kernel_launch(void* const* d_in, const int* in_sizes, int n_in,
                              void* d_out, int out_size, void* d_ws, size_t ws_size,
                              hipStream_t stream) {
    (void)in_sizes; (void)n_in; (void)out_size; (void)d_ws; (void)ws_size;

    const float* x  = (const float*)d_in[0];
    float* ll = (float*)d_out;                                  // first output, flat
    float* hs = ll + (size_t)NIMG * OH * OW;                    // second output, flat

    const int n_threads = NIMG * OH * (OW / 4);                 // 8,388,608
    const int block = 256;                                      // 8 wave32s
    const int grid  = n_threads / block;                        // exact, no tail

    haar_dwt2_kernel<<<grid, block, 0, stream>>>(x, ll, hs);
}